// ContextQueryAttention_75505525063898
// MI455X (gfx1250) — compile-verified
//
#include <hip/hip_runtime.h>

// Problem dimensions (fixed by the reference).
#define B_  8
#define LC  2048
#define LQ  512
#define DD  512
#define NEG_BIAS (-1.0e9f)

typedef float v2f __attribute__((ext_vector_type(2)));
typedef float v8f __attribute__((ext_vector_type(8)));

// CDNA5 fp32 matrix core: D(16x16,f32) = A(16x4,f32) * B(4x16,f32) + C
__device__ __forceinline__ v8f wmma_f32(v2f a, v2f b, v8f c) {
    return __builtin_amdgcn_wmma_f32_16x16x4_f32(
        /*neg_a=*/false, a, /*neg_b=*/false, b,
        /*c_mod=*/(short)0, c, /*reuse_a=*/false, /*reuse_b=*/false);
}

// ---------------------------------------------------------------------------
// K1/K2: per-row dot product  out[r] = sum_d x[r,d]*w[d]   (one wave per row)
// ---------------------------------------------------------------------------
__global__ void row_dot_kernel(const float* __restrict__ x,
                               const float* __restrict__ w,
                               float* __restrict__ out, int rows, int d) {
    int wave = (blockIdx.x * blockDim.x + threadIdx.x) >> 5;
    int lane = threadIdx.x & 31;
    if (wave >= rows) return;
    const float* xr = x + (size_t)wave * d;
    float s = 0.f;
    for (int i = lane; i < d; i += 32) s += xr[i] * w[i];
#pragma unroll
    for (int off = 16; off; off >>= 1) s += __shfl_xor(s, off, 32);
    if (lane == 0) out[wave] = s;
}

// ---------------------------------------------------------------------------
// K3: QW[b,q,d] = sum_e query[b,q,e] * W[d,e]      (M=LQ, N=DD, K=DD)
// per-wave tile: 32(M) x 64(N): 2 A frags x 4 B frags -> 8 WMMA per K-step
// ---------------------------------------------------------------------------
__global__ void qw_gemm_kernel(const float* __restrict__ query,
                               const float* __restrict__ W,
                               float* __restrict__ QW) {
    int gw   = (blockIdx.x * blockDim.x + threadIdx.x) >> 5;
    int lane = threadIdx.x & 31;
    const int wpb = (LQ / 32) * (DD / 64);
    int b  = gw / wpb;
    int t  = gw % wpb;
    int m0 = (t / (DD / 64)) * 32;   // q tile
    int n0 = (t % (DD / 64)) * 64;   // d strip
    const float* Q = query + (size_t)b * LQ * DD;
    float*       O = QW    + (size_t)b * LQ * DD;
    int lm = lane & 15, hi = lane >> 4;

    v8f acc[2][4] = {};
    for (int k = 0; k < DD; k += 4) {
        v2f a[2];
#pragma unroll
        for (int mm = 0; mm < 2; ++mm) {
            const float* ap = Q + (size_t)(m0 + mm * 16 + lm) * DD + k + 2 * hi;
            a[mm].x = ap[0]; a[mm].y = ap[1];
        }
#pragma unroll
        for (int nn = 0; nn < 4; ++nn) {
            const float* bp = W + (size_t)(n0 + nn * 16 + lm) * DD + k + 2 * hi; // B[k][n]=W[n,k]
            v2f bv; bv.x = bp[0]; bv.y = bp[1];
            acc[0][nn] = wmma_f32(a[0], bv, acc[0][nn]);
            acc[1][nn] = wmma_f32(a[1], bv, acc[1][nn]);
        }
    }
#pragma unroll
    for (int mm = 0; mm < 2; ++mm)
#pragma unroll
        for (int nn = 0; nn < 4; ++nn)
#pragma unroll
            for (int j = 0; j < 8; ++j)
                O[(size_t)(m0 + mm * 16 + j + 8 * hi) * DD + n0 + nn * 16 + lm] =
                    acc[mm][nn][j];
}

// ---------------------------------------------------------------------------
// K4: S[b,c,q] = ctx[b,c,:]·QW[b,q,:] + cw[b,c] + qw[b,q] + mask bias
//     (M=LC, N=LQ, K=DD), per-wave 32x64, fused masked-similarity epilogue
// ---------------------------------------------------------------------------
__global__ void sim_gemm_kernel(const float* __restrict__ Ctx,
                                const float* __restrict__ QW,
                                const float* __restrict__ cw,
                                const float* __restrict__ qv,
                                const int* __restrict__ mask,
                                float* __restrict__ S) {
    int gw   = (blockIdx.x * blockDim.x + threadIdx.x) >> 5;
    int lane = threadIdx.x & 31;
    const int wpb = (LC / 32) * (LQ / 64);
    int b  = gw / wpb;
    int t  = gw % wpb;
    int m0 = (t / (LQ / 64)) * 32;   // c tile
    int n0 = (t % (LQ / 64)) * 64;   // q strip
    const float* A  = Ctx + (size_t)b * LC * DD;
    const float* Bm = QW  + (size_t)b * LQ * DD;
    int lm = lane & 15, hi = lane >> 4;

    v8f acc[2][4] = {};
    for (int k = 0; k < DD; k += 4) {
        v2f a[2];
#pragma unroll
        for (int mm = 0; mm < 2; ++mm) {
            const float* ap = A + (size_t)(m0 + mm * 16 + lm) * DD + k + 2 * hi;
            __builtin_prefetch(ap + 64, 0, 0);   // global_prefetch_b8, 64 floats ahead
            a[mm].x = ap[0]; a[mm].y = ap[1];
        }
#pragma unroll
        for (int nn = 0; nn < 4; ++nn) {
            const float* bp = Bm + (size_t)(n0 + nn * 16 + lm) * DD + k + 2 * hi;
            v2f bv; bv.x = bp[0]; bv.y = bp[1];
            acc[0][nn] = wmma_f32(a[0], bv, acc[0][nn]);
            acc[1][nn] = wmma_f32(a[1], bv, acc[1][nn]);
        }
    }
    const float* cwb = cw + (size_t)b * LC;
    const float* qvb = qv + (size_t)b * LQ;
    const int*   mb  = mask + (size_t)b * LC * LQ;
    float*       Sb  = S    + (size_t)b * LC * LQ;
#pragma unroll
    for (int mm = 0; mm < 2; ++mm)
#pragma unroll
        for (int nn = 0; nn < 4; ++nn) {
            int n = n0 + nn * 16 + lm;
#pragma unroll
            for (int j = 0; j < 8; ++j) {
                int m = m0 + mm * 16 + j + 8 * hi;
                float sv = acc[mm][nn][j] + cwb[m] + qvb[n];
                if (mb[(size_t)m * LQ + n] == 0) sv += NEG_BIAS;
                Sb[(size_t)m * LQ + n] = sv;
            }
        }
}

// ---------------------------------------------------------------------------
// K5: row softmax over q (length LQ=512). One wave per row, 16 elems/lane.
// ---------------------------------------------------------------------------
__global__ void row_softmax_kernel(const float* __restrict__ S,
                                   float* __restrict__ P) {
    int row  = (blockIdx.x * blockDim.x + threadIdx.x) >> 5;
    int lane = threadIdx.x & 31;
    const float* Sr = S + (size_t)row * LQ;
    float v[LQ / 32];
    float mx = -1e30f;
#pragma unroll
    for (int i = 0; i < LQ / 32; ++i) { v[i] = Sr[lane + 32 * i]; mx = fmaxf(mx, v[i]); }
#pragma unroll
    for (int off = 16; off; off >>= 1) mx = fmaxf(mx, __shfl_xor(mx, off, 32));
    float s = 0.f;
#pragma unroll
    for (int i = 0; i < LQ / 32; ++i) { v[i] = __expf(v[i] - mx); s += v[i]; }
#pragma unroll
    for (int off = 16; off; off >>= 1) s += __shfl_xor(s, off, 32);
    float inv = 1.0f / s;
    float* Pr = P + (size_t)row * LQ;
#pragma unroll
    for (int i = 0; i < LQ / 32; ++i) Pr[lane + 32 * i] = v[i] * inv;
}

// ---------------------------------------------------------------------------
// K6a: online max/sum over the context axis for each (b,q) column.
// ---------------------------------------------------------------------------
__global__ void col_stats_kernel(const float* __restrict__ S,
                                 float* __restrict__ cmax,
                                 float* __restrict__ csum) {
    int col = blockIdx.x * blockDim.x + threadIdx.x;   // b*LQ + q
    int b = col / LQ, q = col % LQ;
    const float* Sb = S + (size_t)b * LC * LQ + q;
    float m = -1e30f, s = 0.f;
    for (int c = 0; c < LC; ++c) {
        float x = Sb[(size_t)c * LQ];
        if (x > m) { s = s * __expf(m - x) + 1.f; m = x; }
        else       { s += __expf(x - m); }
    }
    cmax[col] = m; csum[col] = s;
}

// K6b: q2c[b,c,q] = exp(S - cmax)/csum, written in place over S.
__global__ void q2c_kernel(float* __restrict__ S,
                           const float* __restrict__ cmax,
                           const float* __restrict__ csum) {
    size_t idx = (size_t)blockIdx.x * blockDim.x + threadIdx.x;
    size_t bi  = idx / ((size_t)LC * LQ);
    int    q   = (int)(idx % LQ);
    int    col = (int)bi * LQ + q;
    S[idx] = __expf(S[idx] - cmax[col]) / csum[col];
}

// ---------------------------------------------------------------------------
// K7: T[b,q,d] = sum_c q2c[b,c,q] * ctx[b,c,d]   (M=LQ, N=DD, K=LC), 32x64
// ---------------------------------------------------------------------------
__global__ void t_gemm_kernel(const float* __restrict__ Q2C,
                              const float* __restrict__ Ctx,
                              float* __restrict__ T) {
    int gw   = (blockIdx.x * blockDim.x + threadIdx.x) >> 5;
    int lane = threadIdx.x & 31;
    const int wpb = (LQ / 32) * (DD / 64);
    int b  = gw / wpb;
    int t  = gw % wpb;
    int m0 = (t / (DD / 64)) * 32;   // q tile
    int n0 = (t % (DD / 64)) * 64;   // d strip
    const float* Pb = Q2C + (size_t)b * LC * LQ;
    const float* Cb = Ctx + (size_t)b * LC * DD;
    int lm = lane & 15, hi = lane >> 4;

    v8f acc[2][4] = {};
    for (int k = 0; k < LC; k += 4) {
        int k0 = k + 2 * hi;
        v2f a[2];                                 // A[m][k] = q2c[k, m] (transposed read)
#pragma unroll
        for (int mm = 0; mm < 2; ++mm) {
            a[mm].x = Pb[(size_t)k0 * LQ + m0 + mm * 16 + lm];
            a[mm].y = Pb[(size_t)(k0 + 1) * LQ + m0 + mm * 16 + lm];
        }
#pragma unroll
        for (int nn = 0; nn < 4; ++nn) {
            int n = n0 + nn * 16 + lm;
            v2f bv;
            bv.x = Cb[(size_t)k0 * DD + n];
            bv.y = Cb[(size_t)(k0 + 1) * DD + n];
            acc[0][nn] = wmma_f32(a[0], bv, acc[0][nn]);
            acc[1][nn] = wmma_f32(a[1], bv, acc[1][nn]);
        }
    }
    float* Tb = T + (size_t)b * LQ * DD;
#pragma unroll
    for (int mm = 0; mm < 2; ++mm)
#pragma unroll
        for (int nn = 0; nn < 4; ++nn)
#pragma unroll
            for (int j = 0; j < 8; ++j)
                Tb[(size_t)(m0 + mm * 16 + j + 8 * hi) * DD + n0 + nn * 16 + lm] =
                    acc[mm][nn][j];
}

// ---------------------------------------------------------------------------
// K8: fused dual GEMM + epilogue, per-wave 32(M) x 32(N), 8 WMMA per K-step.
//  C2Q = c2q @ query,  Q2C = c2q @ T   (M=LC, N=DD, K=LQ); A fragments shared.
//  out[b,c,:] = [ctx | C2Q | ctx*C2Q | ctx*Q2C]
// ---------------------------------------------------------------------------
__global__ void out_gemm_kernel(const float* __restrict__ P,
                                const float* __restrict__ Qry,
                                const float* __restrict__ T,
                                const float* __restrict__ Ctx,
                                float* __restrict__ Out) {
    int gw   = (blockIdx.x * blockDim.x + threadIdx.x) >> 5;
    int lane = threadIdx.x & 31;
    const int wpb = (LC / 32) * (DD / 32);
    int b  = gw / wpb;
    int t  = gw % wpb;
    int m0 = (t / (DD / 32)) * 32;   // c tile
    int n0 = (t % (DD / 32)) * 32;   // d strip
    const float* Pb = P   + (size_t)b * LC * LQ;
    const float* Qb = Qry + (size_t)b * LQ * DD;
    const float* Tb = T   + (size_t)b * LQ * DD;
    int lm = lane & 15, hi = lane >> 4;

    v8f accQ[2][2] = {}, accT[2][2] = {};
    for (int k = 0; k < LQ; k += 4) {
        int k0 = k + 2 * hi;
        v2f a[2];
#pragma unroll
        for (int mm = 0; mm < 2; ++mm) {
            const float* ap = Pb + (size_t)(m0 + mm * 16 + lm) * LQ + k0;
            __builtin_prefetch(ap + 64, 0, 0);
            a[mm].x = ap[0]; a[mm].y = ap[1];
        }
#pragma unroll
        for (int nn = 0; nn < 2; ++nn) {
            int n = n0 + nn * 16 + lm;
            v2f bq, bt;
            bq.x = Qb[(size_t)k0 * DD + n]; bq.y = Qb[(size_t)(k0 + 1) * DD + n];
            bt.x = Tb[(size_t)k0 * DD + n]; bt.y = Tb[(size_t)(k0 + 1) * DD + n];
#pragma unroll
            for (int mm = 0; mm < 2; ++mm) {
                accQ[mm][nn] = wmma_f32(a[mm], bq, accQ[mm][nn]);
                accT[mm][nn] = wmma_f32(a[mm], bt, accT[mm][nn]);
            }
        }
    }
    const float* Cb = Ctx + (size_t)b * LC * DD;
#pragma unroll
    for (int mm = 0; mm < 2; ++mm)
#pragma unroll
        for (int nn = 0; nn < 2; ++nn) {
            int n = n0 + nn * 16 + lm;
#pragma unroll
            for (int j = 0; j < 8; ++j) {
                int m = m0 + mm * 16 + j + 8 * hi;
                float ctx = Cb[(size_t)m * DD + n];
                float c2q = accQ[mm][nn][j];
                float q2c = accT[mm][nn][j];
                float* orow = Out + (size_t)(b * LC + m) * (4 * DD);
                orow[n]          = ctx;
                orow[DD + n]     = c2q;
                orow[2 * DD + n] = ctx * c2q;
                orow[3 * DD + n] = ctx * q2c;
            }
        }
}

// ---------------------------------------------------------------------------
extern "C" void kernel_launch(void* const* d_in, const int* in_sizes, int n_in,
                              void* d_out, int out_size, void* d_ws, size_t ws_size,
                              hipStream_t stream) {
    const float* context = (const float*)d_in[0];
    const float* query   = (const float*)d_in[1];
    const int*   mask    = (const int*)d_in[2];
    const float* query_w = (const float*)d_in[3];
    const float* ctx_w   = (const float*)d_in[4];
    const float* dot_w   = (const float*)d_in[5];
    float* out = (float*)d_out;

    float* ws   = (float*)d_ws;
    float* QW   = ws;                               // B*LQ*DD   (8 MB)
    float* S    = QW   + (size_t)B_ * LQ * DD;      // B*LC*LQ   (32 MB) -> later q2c
    float* P    = S    + (size_t)B_ * LC * LQ;      // B*LC*LQ   (32 MB) c2q
    float* T    = P    + (size_t)B_ * LC * LQ;      // B*LQ*DD   (8 MB)
    float* cw   = T    + (size_t)B_ * LQ * DD;      // B*LC
    float* qv   = cw   + (size_t)B_ * LC;           // B*LQ
    float* cmax = qv   + (size_t)B_ * LQ;           // B*LQ
    float* csum = cmax + (size_t)B_ * LQ;           // B*LQ

    dim3 blk(256);  // 8 wave32s

    // K1/K2: bias terms
    row_dot_kernel<<<(B_ * LC) / 8, blk, 0, stream>>>(context, ctx_w, cw, B_ * LC, DD);
    row_dot_kernel<<<(B_ * LQ) / 8, blk, 0, stream>>>(query, query_w, qv, B_ * LQ, DD);
    // K3: QW = query @ dot_w^T
    qw_gemm_kernel<<<(B_ * (LQ / 32) * (DD / 64)) / 8, blk, 0, stream>>>(query, dot_w, QW);
    // K4: masked similarity
    sim_gemm_kernel<<<(B_ * (LC / 32) * (LQ / 64)) / 8, blk, 0, stream>>>(context, QW, cw, qv, mask, S);
    // K5: c2q = softmax over q
    row_softmax_kernel<<<(B_ * LC) / 8, blk, 0, stream>>>(S, P);
    // K6: q2c = softmax over c (stats, then in-place write over S)
    col_stats_kernel<<<(B_ * LQ) / 256, blk, 0, stream>>>(S, cmax, csum);
    q2c_kernel<<<(int)(((size_t)B_ * LC * LQ) / 256), blk, 0, stream>>>(S, cmax, csum);
    // K7: T = q2c^T @ context
    t_gemm_kernel<<<(B_ * (LQ / 32) * (DD / 64)) / 8, blk, 0, stream>>>(S, context, T);
    // K8: dual GEMM + concat epilogue
    out_gemm_kernel<<<(B_ * (LC / 32) * (DD / 32)) / 8, blk, 0, stream>>>(P, query, T, context, out);
}